// FeatureAlignmentModule_24489903522350
// MI455X (gfx1250) — compile-verified
//
#include <hip/hip_runtime.h>
#include <hip/hip_bf16.h>

typedef __attribute__((ext_vector_type(2))) float v2f;
typedef __attribute__((ext_vector_type(8))) float v8f;

#define DIM_B 16
#define DIM_C 128
#define DIM_H 64
#define DIM_W 64
#define HWSZ  (DIM_H * DIM_W)
#define KOFF  2304   // 256 in-ch * 9 taps (offset conv reduction dim)
#define KDEF  1152   // 128 in-ch * 9 taps (deform einsum reduction dim)
#define NOFF  18
#define K1CH  144    // kernel-1 K chunk  (16 ch * 9), LDS stride 146
#define K2CH  576    // kernel-2 K chunk  (64 ch * 9), LDS stride 578

// -------- Kernel 1: offset = relu(conv3x3(concat(vis, ir), w_off) + b_off) --------
// Implicit GEMM, M = B*H*W, N = 18 (padded 32), K = 2304.
// Block = 256 thr = 8 waves: 4 M-subtiles (64-pixel row) x 2 N-tiles.
// A chunk (64 x 144) staged in LDS with coalesced row loads; GEMM loop is
// ds_load_b64 (A) + contiguous global_load_b64 (B) + V_WMMA_F32_16X16X4_F32.

__global__ __launch_bounds__(256) void offset_conv_wmma(
    const float* __restrict__ vis, const float* __restrict__ ir,
    const float* __restrict__ w_off, const float* __restrict__ b_off,
    float* __restrict__ off_out)
{
  __shared__ float sA[64][K1CH + 2];     // stride 146: even (b64-aligned), no bank conflicts

  const int tid  = threadIdx.x;
  const int lane = tid & 31;
  const int row  = lane & 15;
  const int hi   = lane >> 4;
  const int wv   = tid >> 5;
  const int msub = wv & 3;               // M-subtile 0..3 (16 pixels each)
  const int nt   = wv >> 2;              // N-tile 0..1

  const int m0 = blockIdx.x * 64;        // one full (b, y) image row
  const int b  = m0 >> 12;
  const int y  = (m0 >> 6) & 63;

  const int n = nt * 16 + row;
  const float* wrow = w_off + (size_t)((n < NOFF) ? n : 0) * KOFF;

  v8f acc = {};
  for (int cg = 0; cg < KOFF / K1CH; ++cg) {     // 16 channel-groups
    __syncthreads();                              // protect sA from previous reads
    const int c0 = cg * 16;
    // Stage A chunk: 64 pixels x 144 K-values, coalesced along x.
    for (int e = tid; e < 64 * K1CH; e += 256) {
      const int p  = e & 63;
      const int ct = e >> 6;                      // local K index = cl*9 + t
      const int cl = ct / 9;
      const int t  = ct - 9 * cl;
      const int c  = c0 + cl;                     // concat channel
      const int ty = t / 3;
      const int tx = t - 3 * ty;
      const int gy = y + ty - 1;
      const int gx = p + tx - 1;
      const bool ok = (gy >= 0) & (gy < DIM_H) & (gx >= 0) & (gx < DIM_W);
      const float* src = (c < DIM_C) ? vis : ir;
      const int cc = c & (DIM_C - 1);
      float v = 0.0f;
      if (ok) v = src[(((size_t)b * DIM_C + cc) * DIM_H + gy) * DIM_W + gx];
      sA[p][ct] = v;
    }
    __syncthreads();

    const float* wch = wrow + cg * K1CH;
    #pragma unroll 4
    for (int k = 0; k < K1CH; k += 4) {
      const int ka = k + 2 * hi;
      v2f a  = *(const v2f*)&sA[msub * 16 + row][ka];
      v2f bb = *(const v2f*)(wch + ka);
      acc = __builtin_amdgcn_wmma_f32_16x16x4_f32(
          false, a, false, bb, (short)0, acc, false, false);
    }
  }

  if (n < NOFF) {
    const float bias = b_off[n];
    float* obase = off_out + (((size_t)b * NOFF + n) * DIM_H + y) * DIM_W + msub * 16;
    #pragma unroll
    for (int r = 0; r < 8; ++r) {
      float v = acc[r] + bias;
      v = v > 0.0f ? v : 0.0f;                    // ReLU
      obase[r + 8 * hi] = v;                      // C/D layout: M = r + 8*hi
    }
  }
}

// -------- Kernel 2: deformable conv (bilinear gather + GEMM vs w_def) --------
// Block = 256 thr = 8 waves; 32-pixel M-tile, waves cover all 128 out-ch with
// TWO accumulators each (M rows 0-15 and 16-31) -> one B fragment feeds 2 WMMAs,
// halving L2 weight traffic. A-tile staged in LDS in 576-wide K chunks.

__global__ __launch_bounds__(256) void deform_conv_wmma(
    const float* __restrict__ ir, const float* __restrict__ off_in,
    const float* __restrict__ w_def, const float* __restrict__ b_def,
    float* __restrict__ out)
{
  __shared__ float sval[32][K2CH + 2];   // stride 578: even, bank = 2*row + col
  __shared__ int   sidx[288][4];         // clamped gather indices per (pixel, tap)
  __shared__ float swt [288][4];         // bilinear weights, validity folded in

  const int tid = threadIdx.x;
  const int m0 = blockIdx.x * 32;
  const int b  = m0 >> 12;
  const int y  = (m0 >> 6) & 63;
  const int x0 = m0 & 63;                // 0 or 32

  // Phase 1: per-(pixel, tap) sampling parameters (288 combos)
  for (int i = tid; i < 288; i += 256) {
    const int p = i / 9;
    const int t = i - 9 * p;
    const int xx = x0 + p;
    const float dy = off_in[(((size_t)b * NOFF + 2 * t    ) * DIM_H + y) * DIM_W + xx];
    const float dx = off_in[(((size_t)b * NOFF + 2 * t + 1) * DIM_H + y) * DIM_W + xx];
    const int ty = t / 3;
    const int tx = t - 3 * ty;
    const float fy = dy + (float)(y  + ty - 1);
    const float fx = dx + (float)(xx + tx - 1);
    const float fy0 = floorf(fy);
    const float fx0 = floorf(fx);
    const float wy = fy - fy0;
    const float wx = fx - fx0;
    const int y0i = (int)fy0, x0i = (int)fx0;
    const int y1i = y0i + 1,  x1i = x0i + 1;
    const float wts[4] = { (1.0f - wy) * (1.0f - wx), (1.0f - wy) * wx,
                           wy * (1.0f - wx),          wy * wx };
    const int ys[4] = { y0i, y0i, y1i, y1i };
    const int xs[4] = { x0i, x1i, x0i, x1i };
    #pragma unroll
    for (int j = 0; j < 4; ++j) {
      const int yi = ys[j], xi = xs[j];
      const bool ok = (yi >= 0) & (yi < DIM_H) & (xi >= 0) & (xi < DIM_W);
      int yc = yi < 0 ? 0 : (yi > DIM_H - 1 ? DIM_H - 1 : yi);
      int xc = xi < 0 ? 0 : (xi > DIM_W - 1 ? DIM_W - 1 : xi);
      sidx[i][j] = yc * DIM_W + xc;
      swt [i][j] = ok ? wts[j] : 0.0f;
    }
  }
  __syncthreads();

  const int lane = tid & 31;
  const int row  = lane & 15;
  const int hi   = lane >> 4;
  const int o    = (tid >> 5) * 16 + row;          // output channel (B col)
  const float* wrow = w_def + (size_t)o * KDEF;

  v8f acc0 = {}, acc1 = {};
  for (int ch = 0; ch < KDEF; ch += K2CH) {        // 2 chunks
    const int cbase = ch / 9;                      // 0 or 64
    // Phase 2: stage bilinear-sampled A chunk (32 pixels x 576) into LDS
    for (int e = tid; e < 32 * K2CH; e += 256) {
      const int p  = e & 31;
      const int ct = e >> 5;                       // local K index = cl*9 + t
      const int cl = ct / 9;
      const int t  = ct - 9 * cl;
      const int pt = p * 9 + t;
      const float* base = ir + ((size_t)b * DIM_C + (cbase + cl)) * HWSZ;
      float g = swt[pt][0] * base[sidx[pt][0]]
              + swt[pt][1] * base[sidx[pt][1]]
              + swt[pt][2] * base[sidx[pt][2]]
              + swt[pt][3] * base[sidx[pt][3]];
      sval[p][ct] = g;
    }
    __syncthreads();

    // Phase 3: GEMM over this K chunk; one B fragment -> two WMMAs
    #pragma unroll 4
    for (int k = 0; k < K2CH; k += 4) {
      const int ka = k + 2 * hi;
      v2f a0 = *(const v2f*)&sval[row][ka];
      v2f a1 = *(const v2f*)&sval[16 + row][ka];
      v2f bb = *(const v2f*)(wrow + ch + ka);
      acc0 = __builtin_amdgcn_wmma_f32_16x16x4_f32(
          false, a0, false, bb, (short)0, acc0, false, false);
      acc1 = __builtin_amdgcn_wmma_f32_16x16x4_f32(
          false, a1, false, bb, (short)0, acc1, false, false);
    }
    __syncthreads();
  }

  const float bias = b_def[o];
  float* obase = out + (((size_t)b * DIM_C + o) * DIM_H + y) * DIM_W + x0;
  #pragma unroll
  for (int r = 0; r < 8; ++r) {
    obase[r + 8 * hi]      = acc0[r] + bias;       // M rows 0..15
    obase[16 + r + 8 * hi] = acc1[r] + bias;       // M rows 16..31
  }
}

extern "C" void kernel_launch(void* const* d_in, const int* in_sizes, int n_in,
                              void* d_out, int out_size, void* d_ws, size_t ws_size,
                              hipStream_t stream) {
  const float* vis   = (const float*)d_in[0];
  const float* ir    = (const float*)d_in[1];
  const float* w_off = (const float*)d_in[2];
  const float* b_off = (const float*)d_in[3];
  const float* w_def = (const float*)d_in[4];
  const float* b_def = (const float*)d_in[5];
  float* out = (float*)d_out;
  float* off = (float*)d_ws;   // 16*18*4096 floats = 4.7 MB scratch, fully rewritten

  offset_conv_wmma<<<DIM_B * HWSZ / 64, 256, 0, stream>>>(vis, ir, w_off, b_off, off);
  deform_conv_wmma<<<DIM_B * HWSZ / 32, 256, 0, stream>>>(ir, off, w_def, b_def, out);
}